// MultiResolutionHashEncoding_21629455302887
// MI455X (gfx1250) — compile-verified
//
#include <hip/hip_runtime.h>
#include <math.h>

// Multi-resolution hash-grid encoding (Instant-NGP style) for MI455X (gfx1250).
//
// Gather/bandwidth-bound kernel: 64 MB of hash tables are L2-resident (192 MB L2),
// the 256 MB fp32 output is streamed with non-temporal stores so it does not
// evict the tables. One lane per point, 16 levels fully unrolled, 8 float2
// gathers per level issued as a batch of global_load_b64.

#define NLEVELS   16
#define HASH_SIZE (1u << 19)          // 524288 entries per level
#define HASH_MASK (HASH_SIZE - 1u)
#define P0 73856093u
#define P1 19349663u
#define P2 83492791u

typedef __attribute__((ext_vector_type(2))) float v2f;
typedef __attribute__((ext_vector_type(4))) float v4f;

struct Res16 { int r[NLEVELS]; };

__global__ __launch_bounds__(256) void hashgrid_encode_kernel(
    const float* __restrict__ x,        // [N,3]
    const float* __restrict__ tables,   // [16, 524288, 2]
    float* __restrict__ out,            // [N,32]
    Res16 res, int n)
{
    const int i = blockIdx.x * blockDim.x + threadIdx.x;
    if (i >= n) return;

    const size_t ib = (size_t)i * 3;
    const float px = x[ib + 0];
    const float py = x[ib + 1];
    const float pz = x[ib + 2];

    float acc[2 * NLEVELS];

#pragma unroll
    for (int l = 0; l < NLEVELS; ++l) {
        const int   ri = res.r[l];
        const float rf = (float)ri;

        // scale / floor / frac (frac from pre-clip floor, matching reference)
        const float sx = px * rf, sy = py * rf, sz = pz * rf;
        const float fx = floorf(sx), fy = floorf(sy), fz = floorf(sz);
        const float tx = sx - fx,  ty = sy - fy,  tz = sz - fz;

        // clamp integer base to [0, res-1]
        int bx = (int)fx; bx = bx < 0 ? 0 : (bx > ri - 1 ? ri - 1 : bx);
        int by = (int)fy; by = by < 0 ? 0 : (by > ri - 1 ? ri - 1 : by);
        int bz = (int)fz; bz = bz < 0 ? 0 : (bz > ri - 1 ? ri - 1 : bz);

        // incremental spatial hash: h(c+1) = h(c) + PRIME (uint32 wraparound)
        const unsigned hx0 = (unsigned)bx * P0, hx1 = hx0 + P0;
        const unsigned hy0 = (unsigned)by * P1, hy1 = hy0 + P1;
        const unsigned hz0 = (unsigned)bz * P2, hz1 = hz0 + P2;

        const v2f* __restrict__ tbl =
            reinterpret_cast<const v2f*>(tables) + (size_t)l * HASH_SIZE;

        // 8 corner gathers (corner order = (dx,dy,dz) nested loop of reference)
        v2f f000 = tbl[(hx0 ^ hy0 ^ hz0) & HASH_MASK];
        v2f f001 = tbl[(hx0 ^ hy0 ^ hz1) & HASH_MASK];
        v2f f010 = tbl[(hx0 ^ hy1 ^ hz0) & HASH_MASK];
        v2f f011 = tbl[(hx0 ^ hy1 ^ hz1) & HASH_MASK];
        v2f f100 = tbl[(hx1 ^ hy0 ^ hz0) & HASH_MASK];
        v2f f101 = tbl[(hx1 ^ hy0 ^ hz1) & HASH_MASK];
        v2f f110 = tbl[(hx1 ^ hy1 ^ hz0) & HASH_MASK];
        v2f f111 = tbl[(hx1 ^ hy1 ^ hz1) & HASH_MASK];

        // trilinear weights, (wx*wy)*wz product order as in reference
        const float wx0 = 1.0f - tx, wx1 = tx;
        const float wy0 = 1.0f - ty, wy1 = ty;
        const float wz0 = 1.0f - tz, wz1 = tz;
        const float w00 = wx0 * wy0, w01 = wx0 * wy1;
        const float w10 = wx1 * wy0, w11 = wx1 * wy1;

        const float w000 = w00 * wz0, w001 = w00 * wz1;
        const float w010 = w01 * wz0, w011 = w01 * wz1;
        const float w100 = w10 * wz0, w101 = w10 * wz1;
        const float w110 = w11 * wz0, w111 = w11 * wz1;

        float a0 = 0.0f, a1 = 0.0f;
        a0 = fmaf(w000, f000.x, a0);  a1 = fmaf(w000, f000.y, a1);
        a0 = fmaf(w001, f001.x, a0);  a1 = fmaf(w001, f001.y, a1);
        a0 = fmaf(w010, f010.x, a0);  a1 = fmaf(w010, f010.y, a1);
        a0 = fmaf(w011, f011.x, a0);  a1 = fmaf(w011, f011.y, a1);
        a0 = fmaf(w100, f100.x, a0);  a1 = fmaf(w100, f100.y, a1);
        a0 = fmaf(w101, f101.x, a0);  a1 = fmaf(w101, f101.y, a1);
        a0 = fmaf(w110, f110.x, a0);  a1 = fmaf(w110, f110.y, a1);
        a0 = fmaf(w111, f111.x, a0);  a1 = fmaf(w111, f111.y, a1);

        acc[2 * l + 0] = a0;
        acc[2 * l + 1] = a1;
    }

    // streaming (non-temporal) output: 8 x b128 stores, keeps tables hot in L2
    v4f* op = reinterpret_cast<v4f*>(out + (size_t)i * (2 * NLEVELS));
#pragma unroll
    for (int k = 0; k < 8; ++k) {
        v4f v;
        v.x = acc[4 * k + 0];
        v.y = acc[4 * k + 1];
        v.z = acc[4 * k + 2];
        v.w = acc[4 * k + 3];
        __builtin_nontemporal_store(v, op + k);
    }
}

extern "C" void kernel_launch(void* const* d_in, const int* in_sizes, int n_in,
                              void* d_out, int out_size, void* d_ws, size_t ws_size,
                              hipStream_t stream) {
    const float* x      = (const float*)d_in[0];   // [N,3] fp32
    const float* tables = (const float*)d_in[1];   // [16,524288,2] fp32
    float*       out    = (float*)d_out;           // [N,32] fp32

    const int n = in_sizes[0] / 3;

    // Replicate the reference's resolution table in double precision on host:
    // b = exp((log(2048)-log(16))/15); res_l = floor(16 * b**l)
    Res16 res;
    const double b = exp((log(2048.0) - log(16.0)) / 15.0);
    for (int l = 0; l < NLEVELS; ++l)
        res.r[l] = (int)floor(16.0 * pow(b, (double)l));

    const int block = 256;
    const int grid  = (n + block - 1) / block;
    hashgrid_encode_kernel<<<grid, block, 0, stream>>>(x, tables, out, res, n);
}